// Model_43671227465951
// MI455X (gfx1250) — compile-verified
//
#include <hip/hip_runtime.h>
#include <hip/hip_bf16.h>
#include <math.h>

// ---------------------------------------------------------------------------
// Model dims (compile-time constants from the reference)
// ---------------------------------------------------------------------------
#define Bz        8
#define Lz        96
#define Nz        32
#define TFz       7
#define D_MODEL   128
#define D_INNER   64
#define D_STATE   16
#define DT_RANK   4
#define D_FF      128
#define PRED_LEN  24
#define Tz        (Lz * Nz)          // 3072  (mamba sequence length)
#define Mz        (Bz * Lz * Nz)     // 24576 (flattened token rows)
#define XPW_N     (DT_RANK + 2 * D_STATE)  // 36

// ---------------------------------------------------------------------------
// WMMA types (gfx1250, wave32): V_WMMA_F32_16X16X4_F32
// ---------------------------------------------------------------------------
typedef __attribute__((ext_vector_type(2))) float v2f;
typedef __attribute__((ext_vector_type(8))) float v8f;

#define ACT_NONE 0
#define ACT_RELU 1
#define ACT_SIGM 2

template <int ACT>
__device__ __forceinline__ float apply_act(float v) {
  if constexpr (ACT == ACT_RELU) return v > 0.f ? v : 0.f;
  if constexpr (ACT == ACT_SIGM) return 1.f / (1.f + expf(-v));
  return v;
}

// ---------------------------------------------------------------------------
// Generic fp32 WMMA GEMM:  C(MxN) = act(A(MxK) @ W(KxN) + bias)
// Block = 128 threads (4 waves). Tile: BM=64, BN=32, BK=32.
// Each wave owns a 16-row strip and two 16x16 N-subtiles.
// A tile row-major, B tile TRANSPOSED in LDS so both A and B fragments are
// single aligned ds_load_b64s (even +2 padding keeps 8B alignment and the
// 34-float stride maps 32 lanes onto 32 distinct LDS banks).
// ---------------------------------------------------------------------------
#define GBM 64
#define GBN 32
#define GBK 32

template <int ACT>
__global__ __launch_bounds__(128)
void gemm_wmma_f32(const float* __restrict__ A, const float* __restrict__ W,
                   const float* __restrict__ bias, float* __restrict__ C,
                   int M, int K, int N) {
  __shared__ float sA [GBM][GBK + 2];   // A[row][k]
  __shared__ float sBt[GBN][GBK + 2];   // B^T[col][k]

  const int tid    = threadIdx.x;
  const int wave   = tid >> 5;
  const int lane   = tid & 31;
  const int laneLo = lane & 15;
  const int laneHi = lane >> 4;
  const int rowBase = blockIdx.x * GBM;
  const int colBase = blockIdx.y * GBN;

  v8f acc0 = {};
  v8f acc1 = {};

  for (int k0 = 0; k0 < K; k0 += GBK) {
    // Stage A tile (64x32 fp32): 2048 elems / 128 threads (coalesced, k-fastest)
    for (int e = tid; e < GBM * GBK; e += 128) {
      int r = e >> 5, c = e & (GBK - 1);
      int gr = rowBase + r, gk = k0 + c;
      sA[r][c] = (gr < M && gk < K) ? A[(size_t)gr * K + gk] : 0.f;
    }
    // Stage B tile transposed (32x32): global reads coalesced (col-fastest),
    // LDS writes stride-34 -> conflict-free across 32 lanes.
    for (int e = tid; e < GBK * GBN; e += 128) {
      int r = e >> 5, c = e & (GBN - 1);
      int gk = k0 + r, gc = colBase + c;
      sBt[c][r] = (gk < K && gc < N) ? W[(size_t)gk * N + gc] : 0.f;
    }
    __syncthreads();

    // Prefetch next A tile rows into cache (gfx1250 global_prefetch_b8)
    if (k0 + GBK < K) {
      int pr = rowBase + (tid >> 1);
      if (pr < M) __builtin_prefetch(&A[(size_t)pr * K + k0 + GBK], 0, 1);
    }

    const int arow = wave * 16 + laneLo;
#pragma unroll
    for (int kk = 0; kk < GBK; kk += 4) {
      const int ks = kk + laneHi * 2;      // even -> 8B-aligned b64 loads
      // A 16x4 frag: row = laneLo, k = 2*laneHi + i  (contiguous pair)
      v2f af = *(const v2f*)&sA[arow][ks];
      // B 4x16 frags: col = laneLo (+16), k = 2*laneHi + i (contiguous pair)
      v2f b0 = *(const v2f*)&sBt[laneLo][ks];
      v2f b1 = *(const v2f*)&sBt[16 + laneLo][ks];
      acc0 = __builtin_amdgcn_wmma_f32_16x16x4_f32(false, af, false, b0,
                                                   (short)0, acc0, false, false);
      acc1 = __builtin_amdgcn_wmma_f32_16x16x4_f32(false, af, false, b1,
                                                   (short)0, acc1, false, false);
    }
    __syncthreads();
  }

  // Epilogue: D vgpr v -> row = v + 8*laneHi, col = laneLo (per 16x16 subtile)
#pragma unroll
  for (int v = 0; v < 8; v++) {
    int row = rowBase + wave * 16 + laneHi * 8 + v;
    if (row >= M) continue;
    int c0 = colBase + laneLo;
    int c1 = colBase + 16 + laneLo;
    if (c0 < N) {
      float r = acc0[v] + (bias ? bias[c0] : 0.f);
      C[(size_t)row * N + c0] = apply_act<ACT>(r);
    }
    if (c1 < N) {
      float r = acc1[v] + (bias ? bias[c1] : 0.f);
      C[(size_t)row * N + c1] = apply_act<ACT>(r);
    }
  }
}

// ---------------------------------------------------------------------------
// Elementwise / small kernels
// ---------------------------------------------------------------------------

// feat[m, 0:8] = [x_enc[m], x_mark[b,l,0..6]]   (N=32 so (b*L+l) = m>>5)
__global__ __launch_bounds__(256)
void feat_kernel(const float* __restrict__ x_enc,
                 const float* __restrict__ x_mark,
                 float* __restrict__ feat) {
  int m = blockIdx.x * blockDim.x + threadIdx.x;
  if (m >= Mz) return;
  int bl = m >> 5;
  float* f = feat + (size_t)m * 8;
  f[0] = x_enc[m];
  const float* xm = x_mark + (size_t)bl * TFz;
#pragma unroll
  for (int j = 0; j < TFz; j++) f[1 + j] = xm[j];
}

// in-place: a[i] *= b[i]
__global__ __launch_bounds__(256)
void mul_kernel(float* __restrict__ a, const float* __restrict__ b, int n) {
  int i = blockIdx.x * blockDim.x + threadIdx.x;
  if (i < n) a[i] *= b[i];
}

// depthwise conv (kernel width 2, causal) + SiLU.
// xc = xz[:, 0:64]; xconv[m,d] = silu(xc[m-1,d]*cw[d][0] + xc[m,d]*cw[d][1] + cb[d])
__global__ __launch_bounds__(256)
void conv_silu_kernel(const float* __restrict__ xz,
                      const float* __restrict__ cw,
                      const float* __restrict__ cb,
                      float* __restrict__ xconv) {
  int i = blockIdx.x * blockDim.x + threadIdx.x;
  if (i >= Mz * D_INNER) return;
  int m = i >> 6, d = i & 63;
  int t = m % Tz;
  float xc = xz[(size_t)m * 128 + d];
  float xp = (t == 0) ? 0.f : xz[(size_t)(m - 1) * 128 + d];
  float pre = xp * cw[d * 2 + 0] + xc * cw[d * 2 + 1] + cb[d];
  xconv[i] = pre / (1.f + expf(-pre));
}

// dt[m,d] = softplus( sum_r dbl[m, r] * dtw[r, d] + dtb[d] ),  r in [0,4)
__global__ __launch_bounds__(256)
void dtproj_kernel(const float* __restrict__ dbl,
                   const float* __restrict__ dtw,
                   const float* __restrict__ dtb,
                   float* __restrict__ dt) {
  int i = blockIdx.x * blockDim.x + threadIdx.x;
  if (i >= Mz * D_INNER) return;
  int m = i >> 6, d = i & 63;
  float s = dtb[d];
#pragma unroll
  for (int r = 0; r < DT_RANK; r++)
    s += dbl[(size_t)m * XPW_N + r] * dtw[r * D_INNER + d];
  dt[i] = (s > 20.f) ? s : log1pf(expf(s));
}

// Selective scan. One 16-lane group per (b,d); lane = state index s.
// h = exp(dt*A)*h + dt*B[s]*xconv ; y[t] = sum_s h*C[s]  (shfl_xor reduce)
__global__ __launch_bounds__(256)
void scan_kernel(const float* __restrict__ dt,
                 const float* __restrict__ dbl,
                 const float* __restrict__ xconv,
                 const float* __restrict__ A_log,
                 float* __restrict__ ys) {
  int gid = blockIdx.x * blockDim.x + threadIdx.x;
  int g = gid >> 4;            // (b,d) group
  int s = gid & 15;            // state index
  if (g >= Bz * D_INNER) return;
  int b = g >> 6;
  int d = g & 63;
  float Ads = -expf(A_log[d * D_STATE + s]);
  float h = 0.f;
  for (int t = 0; t < Tz; t++) {
    int m = b * Tz + t;
    float dtv = dt[(size_t)m * D_INNER + d];
    float xc  = xconv[(size_t)m * D_INNER + d];
    float Bv  = dbl[(size_t)m * XPW_N + DT_RANK + s];
    float Cv  = dbl[(size_t)m * XPW_N + DT_RANK + D_STATE + s];
    float dA = expf(dtv * Ads);
    h = dA * h + dtv * Bv * xc;
    float contrib = h * Cv;
#pragma unroll
    for (int off = 1; off < 16; off <<= 1)
      contrib += __shfl_xor(contrib, off, 16);
    if (s == 0) ys[(size_t)m * D_INNER + d] = contrib;
  }
}

// in-place gating: ys = (ys + xconv*Dp[d]) * silu(z), z = xz[:, 64:128]
__global__ __launch_bounds__(256)
void gate_kernel(float* __restrict__ ys, const float* __restrict__ xz,
                 const float* __restrict__ xconv, const float* __restrict__ Dp) {
  int i = blockIdx.x * blockDim.x + threadIdx.x;
  if (i >= Mz * D_INNER) return;
  int m = i >> 6, d = i & 63;
  float z = xz[(size_t)m * 128 + 64 + d];
  float sil = z / (1.f + expf(-z));
  ys[i] = (ys[i] + xconv[i] * Dp[d]) * sil;
}

// time flip within each batch: out[b,t,d] = in[b,T-1-t,d]
__global__ __launch_bounds__(256)
void flip_kernel(const float* __restrict__ in, float* __restrict__ out) {
  int i = blockIdx.x * blockDim.x + threadIdx.x;
  if (i >= Mz * D_INNER) return;
  int m = i >> 6, d = i & 63;
  int b = m / Tz, t = m % Tz;
  int m2 = b * Tz + (Tz - 1 - t);
  out[i] = in[(size_t)m2 * D_INNER + d];
}

// acc[b,t,d] += in[b,T-1-t,d]
__global__ __launch_bounds__(256)
void flipadd_kernel(float* __restrict__ acc, const float* __restrict__ in) {
  int i = blockIdx.x * blockDim.x + threadIdx.x;
  if (i >= Mz * D_INNER) return;
  int m = i >> 6, d = i & 63;
  int b = m / Tz, t = m % Tz;
  int m2 = b * Tz + (Tz - 1 - t);
  acc[i] += in[(size_t)m2 * D_INNER + d];
}

// LayerNorm over last dim (64), optional residual.  One wave per row.
__global__ __launch_bounds__(256)
void ln_kernel(const float* __restrict__ a, const float* __restrict__ res,
               const float* __restrict__ g, const float* __restrict__ be,
               float* __restrict__ out) {
  int lane = threadIdx.x & 31;
  int row = blockIdx.x * 8 + (threadIdx.x >> 5);
  if (row >= Mz) return;
  size_t base = (size_t)row * 64;
  float x0 = a[base + lane] + (res ? res[base + lane] : 0.f);
  float x1 = a[base + 32 + lane] + (res ? res[base + 32 + lane] : 0.f);
  float sum = x0 + x1;
#pragma unroll
  for (int off = 1; off < 32; off <<= 1) sum += __shfl_xor(sum, off, 32);
  float mean = sum * (1.f / 64.f);
  float d0 = x0 - mean, d1 = x1 - mean;
  float v = d0 * d0 + d1 * d1;
#pragma unroll
  for (int off = 1; off < 32; off <<= 1) v += __shfl_xor(v, off, 32);
  float rs = rsqrtf(v * (1.f / 64.f) + 1e-5f);
  out[base + lane]      = d0 * rs * g[lane]      + be[lane];
  out[base + 32 + lane] = d1 * rs * g[32 + lane] + be[32 + lane];
}

// out[b, p, n] = proj[(b*L + L-1)*N + n, p]
__global__ __launch_bounds__(256)
void gather_kernel(const float* __restrict__ proj, float* __restrict__ out) {
  int i = blockIdx.x * blockDim.x + threadIdx.x;
  if (i >= Bz * PRED_LEN * Nz) return;
  int b = i / (PRED_LEN * Nz);
  int rem = i - b * (PRED_LEN * Nz);
  int p = rem / Nz;
  int n = rem - p * Nz;
  int m = (b * Lz + (Lz - 1)) * Nz + n;
  out[i] = proj[(size_t)m * PRED_LEN + p];
}

// ---------------------------------------------------------------------------
// Host-side orchestration
// ---------------------------------------------------------------------------
struct MambaP {
  const float *in_w, *conv_w, *conv_b, *xp_w, *dtp_w, *dtp_b, *A_log, *D, *out_w;
};

static inline MambaP mamba_from(void* const* din, int b) {
  MambaP p;
  p.in_w   = (const float*)din[b + 0];
  p.conv_w = (const float*)din[b + 1];
  p.conv_b = (const float*)din[b + 2];
  p.xp_w   = (const float*)din[b + 3];
  p.dtp_w  = (const float*)din[b + 4];
  p.dtp_b  = (const float*)din[b + 5];
  p.A_log  = (const float*)din[b + 6];
  p.D      = (const float*)din[b + 7];
  p.out_w  = (const float*)din[b + 8];
  return p;
}

static void gemm(const float* A, const float* W, const float* bias, float* C,
                 int M, int K, int N, int act, hipStream_t s) {
  dim3 grid((M + GBM - 1) / GBM, (N + GBN - 1) / GBN);
  dim3 block(128);
  switch (act) {
    case ACT_RELU: gemm_wmma_f32<ACT_RELU><<<grid, block, 0, s>>>(A, W, bias, C, M, K, N); break;
    case ACT_SIGM: gemm_wmma_f32<ACT_SIGM><<<grid, block, 0, s>>>(A, W, bias, C, M, K, N); break;
    default:       gemm_wmma_f32<ACT_NONE><<<grid, block, 0, s>>>(A, W, bias, C, M, K, N); break;
  }
}

static inline dim3 ew_grid(int n) { return dim3((n + 255) / 256); }

struct Scratch {
  float *x, *xz, *att, *xc, *dbl, *dt, *ys, *nw, *xr, *tmp, *proj;
};

static void run_mamba(const float* x, const MambaP& p, float* out,
                      const Scratch& w, hipStream_t s) {
  const int nE = Mz * D_INNER;
  gemm(x, p.in_w, nullptr, w.xz, Mz, D_INNER, 2 * D_INNER, ACT_NONE, s);
  conv_silu_kernel<<<ew_grid(nE), 256, 0, s>>>(w.xz, p.conv_w, p.conv_b, w.xc);
  gemm(w.xc, p.xp_w, nullptr, w.dbl, Mz, D_INNER, XPW_N, ACT_NONE, s);
  dtproj_kernel<<<ew_grid(nE), 256, 0, s>>>(w.dbl, p.dtp_w, p.dtp_b, w.dt);
  scan_kernel<<<dim3((Bz * D_INNER * D_STATE) / 256), 256, 0, s>>>(
      w.dt, w.dbl, w.xc, p.A_log, w.ys);
  gate_kernel<<<ew_grid(nE), 256, 0, s>>>(w.ys, w.xz, w.xc, p.D);
  gemm(w.ys, p.out_w, nullptr, out, Mz, D_INNER, D_INNER, ACT_NONE, s);
}

extern "C" void kernel_launch(void* const* d_in, const int* in_sizes, int n_in,
                              void* d_out, int out_size, void* d_ws, size_t ws_size,
                              hipStream_t stream) {
  (void)in_sizes; (void)n_in; (void)out_size; (void)ws_size;

  // ---- inputs (setup_inputs() dict order, params depth-first) ----
  const float* x_enc    = (const float*)d_in[0];
  const float* x_mark   = (const float*)d_in[1];
  // d_in[2], d_in[3]: x_dec / x_mark_dec (unused by reference forward)
  const float* emb_w  = (const float*)d_in[4];
  const float* emb_b  = (const float*)d_in[5];
  const float* ca_w1  = (const float*)d_in[6];
  const float* ca_b1  = (const float*)d_in[7];
  const float* ca_w2  = (const float*)d_in[8];
  const float* ca_b2  = (const float*)d_in[9];
  const float* red_w  = (const float*)d_in[10];
  const float* red_b  = (const float*)d_in[11];
  const int L0 = 12, LSTRIDE = 26;  // 9 + 9 + 8 arrays per encoder layer
  const float* enc_ng = (const float*)d_in[L0 + 2 * LSTRIDE + 0];
  const float* enc_nb = (const float*)d_in[L0 + 2 * LSTRIDE + 1];
  const float* proj_w = (const float*)d_in[L0 + 2 * LSTRIDE + 2];
  const float* proj_b = (const float*)d_in[L0 + 2 * LSTRIDE + 3];

  // ---- workspace carve-up (floats). ~78 MB total; resident in L2. ----
  float* ws = (float*)d_ws;
  Scratch w;
  size_t o = 0;
  w.x    = ws + o; o += (size_t)Mz * 64;    // encoder activations
  w.xz   = ws + o; o += (size_t)Mz * 128;   // enc / mamba in-proj
  w.att  = ws + o; o += (size_t)Mz * 128;   // channel attention
  w.xc   = ws + o; o += (size_t)Mz * 64;    // a1 / xconv / ffn-out
  w.dbl  = ws + o; o += (size_t)Mz * 40;    // x-proj (stride 36)
  w.dt   = ws + o; o += (size_t)Mz * 64;
  w.ys   = ws + o; o += (size_t)Mz * 64;    // scan out -> gated y
  w.nw   = ws + o; o += (size_t)Mz * 64;    // new_x accumulator
  w.xr   = ws + o; o += (size_t)Mz * 64;    // flipped sequence
  w.tmp  = ws + o; o += (size_t)Mz * 128;   // feat / ffn hidden / rev-mamba out
  w.proj = ws + o; o += (size_t)Mz * PRED_LEN;

  const int nE = Mz * D_INNER;
  const int nW = Mz * D_MODEL;

  // ---- embedding + channel attention ----
  feat_kernel<<<ew_grid(Mz), 256, 0, stream>>>(x_enc, x_mark, w.tmp);
  gemm(w.tmp, emb_w, emb_b, w.xz, Mz, 1 + TFz, D_MODEL, ACT_NONE, stream);       // enc
  gemm(w.xz, ca_w1, ca_b1, w.xc, Mz, D_MODEL, 64, ACT_RELU, stream);             // a1
  gemm(w.xc, ca_w2, ca_b2, w.att, Mz, 64, D_MODEL, ACT_SIGM, stream);            // att
  mul_kernel<<<ew_grid(nW), 256, 0, stream>>>(w.att, w.xz, nW);                  // enc*att
  gemm(w.att, red_w, red_b, w.x, Mz, D_MODEL, D_INNER, ACT_NONE, stream);        // x

  // ---- encoder layers ----
  for (int l = 0; l < 2; l++) {
    int base = L0 + l * LSTRIDE;
    MambaP mf = mamba_from(d_in, base + 0);
    MambaP mr = mamba_from(d_in, base + 9);
    const float* n1g = (const float*)d_in[base + 18];
    const float* n1b = (const float*)d_in[base + 19];
    const float* c1w = (const float*)d_in[base + 20];
    const float* c1b = (const float*)d_in[base + 21];
    const float* c2w = (const float*)d_in[base + 22];
    const float* c2b = (const float*)d_in[base + 23];
    const float* n2g = (const float*)d_in[base + 24];
    const float* n2b = (const float*)d_in[base + 25];

    run_mamba(w.x, mf, w.nw, w, stream);                                   // new_x
    flip_kernel<<<ew_grid(nE), 256, 0, stream>>>(w.x, w.xr);
    run_mamba(w.xr, mr, w.tmp, w, stream);                                 // rev out
    flipadd_kernel<<<ew_grid(nE), 256, 0, stream>>>(w.nw, w.tmp);
    ln_kernel<<<dim3(Mz / 8), 256, 0, stream>>>(w.x, w.nw, n1g, n1b, w.x); // x=LN(x+new)
    gemm(w.x, c1w, c1b, w.tmp, Mz, D_INNER, D_FF, ACT_RELU, stream);
    gemm(w.tmp, c2w, c2b, w.xc, Mz, D_FF, D_INNER, ACT_NONE, stream);
    ln_kernel<<<dim3(Mz / 8), 256, 0, stream>>>(w.x, w.xc, n2g, n2b, w.x); // x=LN(x+ffn)
  }

  // ---- head ----
  ln_kernel<<<dim3(Mz / 8), 256, 0, stream>>>(w.x, nullptr, enc_ng, enc_nb, w.x);
  gemm(w.x, proj_w, proj_b, w.proj, Mz, D_INNER, PRED_LEN, ACT_NONE, stream);
  gather_kernel<<<ew_grid(Bz * PRED_LEN * Nz), 256, 0, stream>>>(w.proj, (float*)d_out);
}